// GradientLayer_53601191854877
// MI455X (gfx1250) — compile-verified
//
#include <hip/hip_runtime.h>
#include <math.h>

#ifndef __has_builtin
#define __has_builtin(x) 0
#endif

#if __has_builtin(__builtin_amdgcn_global_load_async_to_lds_b128)
#define HAS_ASYNC_LDS 1
#else
#define HAS_ASYNC_LDS 0
#endif

#define AS1 __attribute__((address_space(1)))
#define AS3 __attribute__((address_space(3)))

typedef int v4i __attribute__((ext_vector_type(4)));

#define W 512
#define H 512
#define TILE_ROWS 8
#define LDS_STRIDE 516   // 512 + 4 floats pad
#define NTHREADS 256

__device__ __forceinline__ int reflect511(int c) {
    // reflect-101 on [0, 511]:  -1 -> 1,  512 -> 510
    if (c < 0) return -c;
    if (c > 511) return 1022 - c;
    return c;
}

__global__ __launch_bounds__(NTHREADS)
void scharr_polar_kernel(const float* __restrict__ x,
                         float* __restrict__ out,
                         int batch) {
    __shared__ float lds[10 * LDS_STRIDE];

    const int b     = blockIdx.y;
    const int tileY = blockIdx.x;
    const int r0    = tileY * TILE_ROWS;     // first output row of tile
    const int tid   = threadIdx.x;

    const float* img = x + (size_t)b * (size_t)(W * H);

    // ---- Stage 10 input rows (r0-1 .. r0+8, row-reflected) into LDS ----
    // 10 rows * 128 float4-chunks = 1280 chunks / 256 threads = 5 each.
#pragma unroll
    for (int i = 0; i < 5; ++i) {
        int idx   = tid + NTHREADS * i;      // 0..1279
        int lrow  = idx >> 7;                // 0..9
        int chunk = idx & 127;               // float4 chunk within row
        int grow  = reflect511(r0 - 1 + lrow);
        const float* gsrc = img + (size_t)grow * W + chunk * 4;
        float*       ldst = &lds[lrow * LDS_STRIDE + chunk * 4];
#if HAS_ASYNC_LDS
        __builtin_amdgcn_global_load_async_to_lds_b128(
            (v4i*)gsrc, (AS3 v4i*)ldst, 0, 0);
#else
        float4 v = *(const float4*)gsrc;
        *(float4*)ldst = v;
#endif
    }
#if HAS_ASYNC_LDS
#if __has_builtin(__builtin_amdgcn_s_wait_asynccnt)
    __builtin_amdgcn_s_wait_asynccnt(0);
#else
    asm volatile("s_wait_asynccnt 0" ::: "memory");
#endif
#endif
    __syncthreads();

    // ---- Compute: lane l, chunk j -> float4 of pixels at col 4*(l+32j) ----
    const int lrow = tid >> 5;               // output row within tile: 0..7
    const int lane = tid & 31;
    const int row  = r0 + lrow;              // global output row (< 512)

    const float* l0 = &lds[(lrow + 0) * LDS_STRIDE];  // row-1
    const float* l1 = &lds[(lrow + 1) * LDS_STRIDE];  // row
    const float* l2 = &lds[(lrow + 2) * LDS_STRIDE];  // row+1

    const size_t plane = (size_t)W * (size_t)H;
    const size_t obase = (size_t)b * plane + (size_t)row * W;
    float* vout = out + obase;
    float* aout = out + (size_t)batch * plane + obase;

    const float TWO_PI_F = 6.28318530717958647692f;

#pragma unroll
    for (int j = 0; j < 4; ++j) {
        const int c0 = 4 * (lane + 32 * j);

        // Need cols c0-1 .. c0+4 from 3 rows (column reflect at image edges).
        float t0[6], t1[6], t2[6];
#pragma unroll
        for (int k = 0; k < 6; ++k) {
            int c = reflect511(c0 - 1 + k);
            t0[k] = l0[c];
            t1[k] = l1[c];
            t2[k] = l2[c];
        }

        float vv[4], aa[4];
#pragma unroll
        for (int p = 0; p < 4; ++p) {
            const float a00 = t0[p], a01 = t0[p + 1], a02 = t0[p + 2];
            const float a10 = t1[p],                  a12 = t1[p + 2];
            const float a20 = t2[p], a21 = t2[p + 1], a22 = t2[p + 2];

            // Scharr cross-correlation
            float gx = 3.0f * (a02 - a00) + 10.0f * (a12 - a10) + 3.0f * (a22 - a20);
            float gy = 3.0f * (a20 - a00) + 10.0f * (a21 - a01) + 3.0f * (a22 - a02);

            vv[p] = sqrtf(gx * gx + gy * gy);
            float an = atan2f(gy, gx);
            aa[p] = (an < 0.0f) ? (an + TWO_PI_F) : an;
        }

        float4 v4 = make_float4(vv[0], vv[1], vv[2], vv[3]);
        float4 a4 = make_float4(aa[0], aa[1], aa[2], aa[3]);
        *(float4*)(vout + c0) = v4;   // contiguous 512B per wave per store
        *(float4*)(aout + c0) = a4;
    }
}

extern "C" void kernel_launch(void* const* d_in, const int* in_sizes, int n_in,
                              void* d_out, int out_size, void* d_ws, size_t ws_size,
                              hipStream_t stream) {
    const float* x = (const float*)d_in[0];
    float* out     = (float*)d_out;

    const int batch = in_sizes[0] / (W * H);   // 64 for the reference shapes

    dim3 grid(H / TILE_ROWS, batch);           // 64 x 64 = 4096 blocks
    dim3 block(NTHREADS);
    scharr_polar_kernel<<<grid, block, 0, stream>>>(x, out, batch);
}